// GeometricExtractor_58892591563300
// MI455X (gfx1250) — compile-verified
//
#include <hip/hip_runtime.h>
#include <math.h>

// GeometricExtractor for MI455X (gfx1250, wave32).
// B=8, N=4096, C=3, K=9. Output (B, N, 6) float32.
//
// Pass A: V_WMMA_F32_16X16X4_F32 Gram tiles + branch-free med3 top-10 scores.
// Pass B: bit-exact WMMA recompute + threshold index collection.
// Phase 2: polar-angle sorting network, centroids + normals, mean.

#define BATCH 8
#define NPTS  4096
#define KNEI  9
#define NTILE (NPTS / 16)      // 256 column tiles per row
#define TOPK  10               // K+1 (self included)

typedef __attribute__((ext_vector_type(2))) float v2f;
typedef __attribute__((ext_vector_type(8))) float v8f;

static __device__ __forceinline__ float med3f(float a, float b, float c) {
#if __has_builtin(__builtin_amdgcn_fmed3f)
    return __builtin_amdgcn_fmed3f(a, b, c);
#else
    return fmaxf(fminf(a, b), fminf(fmaxf(a, b), c));
#endif
}

// ---------------------------------------------------------------------------
// Kernel 0: pack [x, y, z, -(x^2+y^2+z^2)] per point into d_ws (float4).
// ---------------------------------------------------------------------------
__global__ void __launch_bounds__(256)
pack_points_kernel(const float* __restrict__ x, float4* __restrict__ pk, int total)
{
    int i = blockIdx.x * blockDim.x + threadIdx.x;
    if (i < total) {
        float a = x[3 * i + 0];
        float b = x[3 * i + 1];
        float c = x[3 * i + 2];
        pk[i] = make_float4(a, b, c, -(a * a + b * b + c * c));
    }
}

// ---------------------------------------------------------------------------
// Kernel 1: one wave (32 threads) per 16-row tile. grid = BATCH * NTILE.
// ---------------------------------------------------------------------------
__global__ void __launch_bounds__(32)
knn_feature_kernel(const float4* __restrict__ pk, float* __restrict__ out)
{
    const int tile    = blockIdx.x;         // [0, BATCH*NTILE)
    const int b       = tile >> 8;          // tile / NTILE
    const int rowTile = tile & (NTILE - 1);
    const int lane    = threadIdx.x;        // 0..31
    const int hl      = lane >> 4;          // half: 0 or 1
    const int l16     = lane & 15;

    const float4* __restrict__ pkb = pk + (size_t)b * NPTS;

    // --- A matrix registers: row point = rowTile*16 + l16 -----------------
    // A layout (32-bit 16x4): lanes 0-15 -> K0 (vgpr0), K1 (vgpr1);
    //                         lanes 16-31 -> K2, K3.  A_i = [2x, 2y, 2z, 1]
    const float4 ap = pkb[rowTile * 16 + l16];
    v2f a;
    a.x = hl ? (2.0f * ap.z) : (2.0f * ap.x);
    a.y = hl ? 1.0f          : (2.0f * ap.y);

    __shared__ float sc_lds[2][32][8];   // score tiles, lane-major
    __shared__ float red[32][TOPK];      // per-lane top-10 score lists
    __shared__ float Tsh[16];            // per-row 10th-largest score
    __shared__ int   gbuf[32][TOPK];     // indices with score >  T
    __shared__ int   ebuf[32][TOPK];     // indices with score == T (ties)
    __shared__ int   cnts[32][2];        // per-lane {g, e} counts
    __shared__ int   selbuf[16][TOPK];   // per-row merged top-10 indices

    const v8f cz = {0.f, 0.f, 0.f, 0.f, 0.f, 0.f, 0.f, 0.f};
    const int srcHalf = (l16 >= 8) ? 16 : 0;
    const int srcV    = l16 & 7;

    // ======================= Pass A: find threshold =======================
    float ts[TOPK];
#pragma unroll
    for (int q = 0; q < TOPK; ++q) ts[q] = -3.4e38f;

    for (int it = 0; it < NTILE / 2; ++it) {
        const int c0 = it * 32 + l16;
        const float4 p0 = pkb[c0];
        const float4 p1 = pkb[c0 + 16];
        {
            int pf = c0 + 32; if (pf > NPTS - 1) pf = NPTS - 1;
            __builtin_prefetch((const void*)(pkb + pf), 0, 0);
        }
        // B layout (4x16), symmetric to A. B_j = [x, y, z, -|x_j|^2]
        v2f b0, b1;
        b0.x = hl ? p0.z : p0.x;  b0.y = hl ? p0.w : p0.y;
        b1.x = hl ? p1.z : p1.x;  b1.y = hl ? p1.w : p1.y;

        v8f d0 = __builtin_amdgcn_wmma_f32_16x16x4_f32(
            false, a, false, b0, (short)0, cz, false, false);
        v8f d1 = __builtin_amdgcn_wmma_f32_16x16x4_f32(
            false, a, false, b1, (short)0, cz, false, false);

#pragma unroll
        for (int v = 0; v < 8; ++v) {
            sc_lds[0][lane][v] = d0[v];
            sc_lds[1][lane][v] = d1[v];
        }
        __syncthreads();

        // Branch-free sorted top-10 update: ts descending.
#pragma unroll
        for (int n = 0; n < 16; ++n) {
            const float cs = sc_lds[hl][srcHalf + n][srcV];
#pragma unroll
            for (int q = TOPK - 1; q >= 1; --q)
                ts[q] = med3f(ts[q - 1], ts[q], cs);   // reads OLD ts[q-1]
            ts[0] = fmaxf(ts[0], cs);
        }
        __syncthreads();
    }

    // Row threshold T = 10th largest of the union of the two half lists.
#pragma unroll
    for (int q = 0; q < TOPK; ++q) red[lane][q] = ts[q];
    __syncthreads();
    if (lane < 16) {
        int pa = 0, pb = 0;
        float T = -3.4e38f;
#pragma unroll
        for (int q = 0; q < TOPK; ++q) {
            const float sa = red[lane][pa];
            const float sb = red[lane + 16][pb];
            if (sa >= sb) { T = sa; ++pa; } else { T = sb; ++pb; }
        }
        Tsh[l16] = T;
    }
    __syncthreads();
    const float T = Tsh[l16];

    // ================= Pass B: bit-exact recompute + collect ==============
    int g = 0, e = 0;
    for (int it = 0; it < NTILE / 2; ++it) {
        const int c0 = it * 32 + l16;
        const float4 p0 = pkb[c0];
        const float4 p1 = pkb[c0 + 16];
        v2f b0, b1;
        b0.x = hl ? p0.z : p0.x;  b0.y = hl ? p0.w : p0.y;
        b1.x = hl ? p1.z : p1.x;  b1.y = hl ? p1.w : p1.y;

        v8f d0 = __builtin_amdgcn_wmma_f32_16x16x4_f32(
            false, a, false, b0, (short)0, cz, false, false);
        v8f d1 = __builtin_amdgcn_wmma_f32_16x16x4_f32(
            false, a, false, b1, (short)0, cz, false, false);

#pragma unroll
        for (int v = 0; v < 8; ++v) {
            sc_lds[0][lane][v] = d0[v];
            sc_lds[1][lane][v] = d1[v];
        }
        __syncthreads();

        const int colbase = it * 32 + hl * 16;
#pragma unroll
        for (int n = 0; n < 16; ++n) {
            const float cs = sc_lds[hl][srcHalf + n][srcV];
            if (cs >= T) {                       // rare path (~10 / 2048)
                const int ci = colbase + n;
                if (cs > T) {
                    if (g < TOPK) { gbuf[lane][g] = ci; ++g; }
                } else if (e < TOPK) {
                    ebuf[lane][e] = ci; ++e;     // ties kept in ascending idx
                }
            }
        }
        __syncthreads();
    }
    cnts[lane][0] = g;
    cnts[lane][1] = e;
    __syncthreads();

    // ===================== Merge halves, drop self ========================
    if (lane < 16) {
        const int ga = cnts[lane][0],      gb = cnts[lane + 16][0];
        const int ea = cnts[lane][1],      eb = cnts[lane + 16][1];
        int t = 0;
        for (int q = 0; q < ga && t < TOPK; ++q) selbuf[l16][t++] = gbuf[lane][q];
        for (int q = 0; q < gb && t < TOPK; ++q) selbuf[l16][t++] = gbuf[lane + 16][q];
        int pa = 0, pb = 0;
        while (t < TOPK) {                   // fill with ties, lowest index first
            const int ia = (pa < ea) ? ebuf[lane][pa]      : 0x7fffffff;
            const int ib = (pb < eb) ? ebuf[lane + 16][pb] : 0x7fffffff;
            if (ia == 0x7fffffff && ib == 0x7fffffff) { selbuf[l16][t++] = 0; continue; }
            if (ia < ib) { selbuf[l16][t++] = ia; ++pa; }
            else         { selbuf[l16][t++] = ib; ++pb; }
        }
        // Remove self (the row max); reuse gbuf row as the final 9 list.
        const int selfIdx = rowTile * 16 + l16;
        int w = 0;
#pragma unroll
        for (int q = 0; q < TOPK; ++q) {
            const int id = selbuf[l16][q];
            if (id != selfIdx && w < KNEI) { gbuf[lane][w] = id; ++w; }
        }
        while (w < KNEI) { gbuf[lane][w] = selbuf[l16][0]; ++w; }  // safety pad

        // ===================== Phase 2: features ==========================
        float rx[KNEI], ry[KNEI], rz[KNEI], phv[KNEI];
#pragma unroll
        for (int q = 0; q < KNEI; ++q) {
            const float4 pn = pkb[gbuf[lane][q]];
            rx[q] = pn.x - ap.x;
            ry[q] = pn.y - ap.y;
            rz[q] = pn.z - ap.z;
            phv[q] = atan2f(ry[q], rx[q]);
        }

        // Sort 9 neighbors by phi ascending: 25-comparator network.
#define CE(i, j)                                                          \
        do {                                                              \
            if (phv[i] > phv[j]) {                                        \
                float t_;                                                 \
                t_ = phv[i]; phv[i] = phv[j]; phv[j] = t_;                \
                t_ = rx[i];  rx[i]  = rx[j];  rx[j]  = t_;                \
                t_ = ry[i];  ry[i]  = ry[j];  ry[j]  = t_;                \
                t_ = rz[i];  rz[i]  = rz[j];  rz[j]  = t_;                \
            }                                                             \
        } while (0)
        CE(0,3); CE(1,7); CE(2,5); CE(4,8);
        CE(0,7); CE(2,4); CE(3,8); CE(5,6);
        CE(0,2); CE(1,3); CE(4,5); CE(7,8);
        CE(1,4); CE(3,6); CE(5,7);
        CE(0,1); CE(2,4); CE(3,5); CE(6,8);
        CE(2,3); CE(4,5); CE(6,7);
        CE(1,2); CE(3,4); CE(5,6);
#undef CE

        float acc0 = 0.f, acc1 = 0.f, acc2 = 0.f;
        float acc3 = 0.f, acc4 = 0.f, acc5 = 0.f;
#pragma unroll
        for (int q = 0; q < KNEI; ++q) {
            const int q2 = (q + 1 == KNEI) ? 0 : q + 1;
            const float v1x = rx[q],  v1y = ry[q],  v1z = rz[q];
            const float v2x = rx[q2], v2y = ry[q2], v2z = rz[q2];
            const float cx  = 0.5f * (v1x + v2x);
            const float cy  = 0.5f * (v1y + v2y);
            const float czl = 0.5f * (v1z + v2z);
            float nx = v1y * v2z - v1z * v2y;
            float ny = v1z * v2x - v1x * v2z;
            float nz = v1x * v2y - v1y * v2x;
            const float nrm = sqrtf(nx * nx + ny * ny + nz * nz) + 1e-6f;
            const float sg  = (czl > 0.0f) ? 1.0f : -1.0f;
            const float inv = sg / nrm;
            acc0 += cx;  acc1 += cy;  acc2 += czl;
            acc3 += nx * inv; acc4 += ny * inv; acc5 += nz * inv;
        }

        const int pt = rowTile * 16 + l16;
        float* __restrict__ o = out + ((size_t)b * NPTS + pt) * 6;
        const float s = 1.0f / 9.0f;
        o[0] = acc0 * s; o[1] = acc1 * s; o[2] = acc2 * s;
        o[3] = acc3 * s; o[4] = acc4 * s; o[5] = acc5 * s;
    }
}

// ---------------------------------------------------------------------------
extern "C" void kernel_launch(void* const* d_in, const int* in_sizes, int n_in,
                              void* d_out, int out_size, void* d_ws, size_t ws_size,
                              hipStream_t stream)
{
    const float* x   = (const float*)d_in[0];  // (B, N, 3) float32
    float4*      pk  = (float4*)d_ws;          // B*N float4 = 512 KB scratch
    float*       out = (float*)d_out;          // (B, N, 6) float32

    const int total = BATCH * NPTS;
    pack_points_kernel<<<(total + 255) / 256, 256, 0, stream>>>(x, pk, total);
    knn_feature_kernel<<<BATCH * NTILE, 32, 0, stream>>>(pk, out);
}